// IOEGCNCLS_33560874451343
// MI455X (gfx1250) — compile-verified
//
#include <hip/hip_runtime.h>
#include <hip/hip_bf16.h>
#include <math.h>

typedef __attribute__((ext_vector_type(2))) float v2f;
typedef __attribute__((ext_vector_type(8))) float v8f;

#define BATCH 1024
#define DIM   512
#define NCLS  18
#define KNN   10
#define QN    32768
#define TEMPI (1.0f / 0.07f)
#define EPSBN 1e-5f

// LDS row stride (floats) for the staged fused tile in knn_partial:
// 516*4 = 2064 B -> 16B aligned rows (b128 stores legal) and 2064/4 % 64 = 4
// banks of skew per row -> all 16 rows land on distinct bank groups, and the
// half-wave (+2 word) offset keeps lanes 16-31 disjoint from lanes 0-15.
#define FSTRIDE 516

// ---------------------------------------------------------------------------
// WMMA f32 GEMM:  C[M,N] (+)= A[M,K] @ W[K,N] (+ bias[N])
// One wave per 16x32 C tile (two 16x16 WMMA accumulators sharing the A frag).
// M mult of 16, N mult of 32, K mult of 4.
// A fragment (16x4 f32): lane l(0-15) holds A[row0+l][k], A[row0+l][k+1];
//                        lane 16+l holds A[row0+l][k+2], A[row0+l][k+3].
// B fragment (4x16 f32): lane l(0-15) holds W[k][col0+l], W[k+1][col0+l];
//                        lane 16+l holds W[k+2][col0+l], W[k+3][col0+l].
// C (16x16 f32): acc[r] = C[row0 + r + 8*(lane>=16)][col0 + lane%16].
// ---------------------------------------------------------------------------
__global__ __launch_bounds__(32)
void gemm_wmma(const float* __restrict__ A, const float* __restrict__ W,
               const float* __restrict__ bias, float* __restrict__ C,
               int M, int N, int Kd, int accumulate)
{
    const int lane = threadIdx.x;
    const int l    = lane & 15;
    const int half = lane >> 4;
    const int col0 = blockIdx.x * 32;
    const int row0 = blockIdx.y * 16;

    v8f acc0, acc1;
    if (accumulate) {
#pragma unroll
        for (int r = 0; r < 8; ++r) {
            const size_t ro = (size_t)(row0 + r + 8 * half) * N;
            acc0[r] = C[ro + col0 + l];
            acc1[r] = C[ro + col0 + 16 + l];
        }
    } else {
        const float b0 = bias ? bias[col0 + l] : 0.0f;
        const float b1 = bias ? bias[col0 + 16 + l] : 0.0f;
#pragma unroll
        for (int r = 0; r < 8; ++r) { acc0[r] = b0; acc1[r] = b1; }
    }

    const float* ap = A + (size_t)(row0 + l) * Kd + 2 * half;
    const float* wp = W + (size_t)(2 * half) * N + col0 + l;

    for (int k = 0; k < Kd; k += 4) {
        v2f a;  a.x  = ap[k];                 a.y  = ap[k + 1];
        v2f b0; b0.x = wp[(size_t)k * N];     b0.y = wp[(size_t)(k + 1) * N];
        v2f b1; b1.x = wp[(size_t)k * N + 16];b1.y = wp[(size_t)(k + 1) * N + 16];
        acc0 = __builtin_amdgcn_wmma_f32_16x16x4_f32(
                   false, a, false, b0, (short)0, acc0, false, false);
        acc1 = __builtin_amdgcn_wmma_f32_16x16x4_f32(
                   false, a, false, b1, (short)0, acc1, false, false);
    }

#pragma unroll
    for (int r = 0; r < 8; ++r) {
        const size_t ro = (size_t)(row0 + r + 8 * half) * N;
        C[ro + col0 + l]      = acc0[r];
        C[ro + col0 + 16 + l] = acc1[r];
    }
}

// ---------------------------------------------------------------------------
// BatchNorm column stats over M rows (biased variance), one block per column.
// ---------------------------------------------------------------------------
__global__ void bn_stats(const float* __restrict__ X, int M, int N,
                         float* __restrict__ mean, float* __restrict__ rstd)
{
    const int j = blockIdx.x;
    float s = 0.0f, s2 = 0.0f;
    for (int i = threadIdx.x; i < M; i += blockDim.x) {
        float v = X[(size_t)i * N + j];
        s += v; s2 += v * v;
    }
    __shared__ float sh[256], sh2[256];
    sh[threadIdx.x] = s; sh2[threadIdx.x] = s2;
    __syncthreads();
    for (int t = blockDim.x >> 1; t > 0; t >>= 1) {
        if ((int)threadIdx.x < t) {
            sh[threadIdx.x]  += sh[threadIdx.x + t];
            sh2[threadIdx.x] += sh2[threadIdx.x + t];
        }
        __syncthreads();
    }
    if (threadIdx.x == 0) {
        float m   = sh[0] / (float)M;
        float var = sh2[0] / (float)M - m * m;
        mean[j] = m;
        rstd[j] = rsqrtf(var + EPSBN);
    }
}

__global__ void bn_apply_relu(const float* __restrict__ X, float* __restrict__ Y,
                              const float* __restrict__ mean, const float* __restrict__ rstd,
                              const float* __restrict__ g, const float* __restrict__ be,
                              int total, int N)
{
    int idx = blockIdx.x * blockDim.x + threadIdx.x;
    if (idx >= total) return;
    int j = idx % N;
    float v = (X[idx] - mean[j]) * rstd[j] * g[j] + be[j];
    Y[idx] = fmaxf(v, 0.0f);
}

// fb[i][j] = fus_b[j] + fus_w[1536+i][j] (i=0..2);  fb[3][j] = fus_b[j] + sum of 3 rows
__global__ void fuse_bias(const float* __restrict__ fus_w, const float* __restrict__ fus_b,
                          float* __restrict__ fb)
{
    int j = blockIdx.x * blockDim.x + threadIdx.x;
    if (j >= DIM) return;
    float b  = fus_b[j];
    float r0 = fus_w[(size_t)1536 * DIM + j];
    float r1 = fus_w[(size_t)1537 * DIM + j];
    float r2 = fus_w[(size_t)1538 * DIM + j];
    fb[0 * DIM + j] = b + r0;
    fb[1 * DIM + j] = b + r1;
    fb[2 * DIM + j] = b + r2;
    fb[3 * DIM + j] = b + r0 + r1 + r2;
}

// small dense layer (D -> NCLS), plain VALU — output is tiny
__global__ void gemm_small(const float* __restrict__ A, const float* __restrict__ W,
                           const float* __restrict__ bias, float* __restrict__ C,
                           int M, int N, int Kd)
{
    int idx = blockIdx.x * blockDim.x + threadIdx.x;
    if (idx >= M * N) return;
    int m = idx / N, n = idx % N;
    float s = bias[n];
    const float* a = A + (size_t)m * Kd;
    for (int k = 0; k < Kd; ++k) s += a[k] * W[(size_t)k * N + n];
    C[(size_t)m * N + n] = s;
}

// per-row squared L2 norm
__global__ void row_sqnorm(const float* __restrict__ X, float* __restrict__ out, int N)
{
    const int r = blockIdx.x;
    float s = 0.0f;
    for (int c = threadIdx.x; c < N; c += blockDim.x) {
        float v = X[(size_t)r * N + c];
        s += v * v;
    }
    __shared__ float sh[128];
    sh[threadIdx.x] = s;
    __syncthreads();
    for (int t = blockDim.x >> 1; t > 0; t >>= 1) {
        if ((int)threadIdx.x < t) sh[threadIdx.x] += sh[threadIdx.x + t];
        __syncthreads();
    }
    if (threadIdx.x == 0) out[r] = sh[0];
}

__device__ __forceinline__ void topk_insert(float* tv, int* ti, float v, int idx)
{
    int w = 0; float wv = tv[0];
#pragma unroll
    for (int k = 1; k < KNN; ++k) if (tv[k] > wv) { wv = tv[k]; w = k; }
    if (v < wv) { tv[w] = v; ti[w] = idx; }
}

// ---------------------------------------------------------------------------
// kNN partial: S^T = queue @ fused^T by 16x16 WMMA tiles.
// grid.x = BATCH/16 (batch column tiles), grid.y = 32 queue splits (1024 rows each).
// The 16 fused rows (32 KB) are staged into LDS once per workgroup; the B
// fragment is then a bank-conflict-free ds_load_b64 while queue streams from
// global. Each lane tracks top-10 smallest distances for ONE batch row;
// lanes l and l+16 (same batch row, different queue rows) merge via LDS.
// ---------------------------------------------------------------------------
__global__ __launch_bounds__(32)
void knn_partial(const float* __restrict__ queue, const float* __restrict__ fused,
                 const float* __restrict__ kn, const float* __restrict__ fn,
                 float* __restrict__ cand_v, int* __restrict__ cand_i)
{
    __shared__ float flds[16 * FSTRIDE];

    const int lane  = threadIdx.x;
    const int l     = lane & 15;
    const int half  = lane >> 4;
    const int col0  = blockIdx.x * 16;       // batch rows covered: col0..col0+15
    const int mycol = col0 + l;
    const float fnv = fn[mycol];
    const int qbase = blockIdx.y * 1024;

    // cooperative stage of fused[col0..col0+15][0..511] -> LDS (float4 copies)
    for (int t = lane; t < 16 * (DIM / 4); t += 32) {
        const int row = t >> 7;              // t / 128
        const int c4  = (t & 127) << 2;      // (t % 128) * 4
        const float4 v = *(const float4*)(fused + (size_t)(col0 + row) * DIM + c4);
        *(float4*)(&flds[row * FSTRIDE + c4]) = v;
    }
    __syncthreads();

    float tv[KNN]; int ti[KNN];
#pragma unroll
    for (int k = 0; k < KNN; ++k) { tv[k] = 3.0e38f; ti[k] = 0; }

    const float* lp = flds + l * FSTRIDE + 2 * half;   // per-lane B-fragment base

    for (int qt = 0; qt < 64; ++qt) {
        const int qrow0 = qbase + qt * 16;
        v8f acc = {};
        const float* ap = queue + (size_t)(qrow0 + l) * DIM + 2 * half;
        for (int k = 0; k < DIM; k += 4) {
            v2f a; a.x = ap[k]; a.y = ap[k + 1];
            v2f b; b.x = lp[k]; b.y = lp[k + 1];
            acc = __builtin_amdgcn_wmma_f32_16x16x4_f32(
                      false, a, false, b, (short)0, acc, false, false);
        }
#pragma unroll
        for (int r = 0; r < 8; ++r) {
            int q = qrow0 + r + 8 * half;
            float d2 = kn[q] + fnv - 2.0f * acc[r];
            topk_insert(tv, ti, sqrtf(fmaxf(d2, 0.0f)), q);
        }
    }

    __shared__ float sv[16][KNN];
    __shared__ int   si[16][KNN];
    if (half) {
#pragma unroll
        for (int k = 0; k < KNN; ++k) { sv[l][k] = tv[k]; si[l][k] = ti[k]; }
    }
    __syncthreads();
    if (!half) {
#pragma unroll
        for (int k = 0; k < KNN; ++k) topk_insert(tv, ti, sv[l][k], si[l][k]);
        const size_t base = (size_t)mycol * (32 * KNN) + blockIdx.y * KNN;
#pragma unroll
        for (int k = 0; k < KNN; ++k) { cand_v[base + k] = tv[k]; cand_i[base + k] = ti[k]; }
    }
}

// merge 32x10 candidates per batch row -> global top-10, softmax(-d/T), weighted sum
__global__ void knn_finalize(const float* __restrict__ cand_v, const int* __restrict__ cand_i,
                             const float* __restrict__ queue, float* __restrict__ neigh)
{
    const int b = blockIdx.x;
    __shared__ float w[KNN];
    __shared__ int   id[KNN];
    if (threadIdx.x == 0) {
        float bv[KNN]; int bi[KNN];
#pragma unroll
        for (int k = 0; k < KNN; ++k) { bv[k] = 3.0e38f; bi[k] = 0; }
        const size_t base = (size_t)b * (32 * KNN);
        for (int c = 0; c < 32 * KNN; ++c)
            topk_insert(bv, bi, cand_v[base + c], cand_i[base + c]);
        float mx = -3.0e38f;
#pragma unroll
        for (int k = 0; k < KNN; ++k) mx = fmaxf(mx, -bv[k] * TEMPI);
        float e[KNN]; float sum = 0.0f;
#pragma unroll
        for (int k = 0; k < KNN; ++k) { e[k] = __expf(-bv[k] * TEMPI - mx); sum += e[k]; }
        float inv = 1.0f / sum;
#pragma unroll
        for (int k = 0; k < KNN; ++k) { w[k] = e[k] * inv; id[k] = bi[k]; }
    }
    __syncthreads();
    for (int c = threadIdx.x; c < DIM; c += blockDim.x) {
        float s = 0.0f;
#pragma unroll
        for (int k = 0; k < KNN; ++k) s += w[k] * queue[(size_t)id[k] * DIM + c];
        neigh[(size_t)b * DIM + c] = s;
    }
}

// ---------------------------------------------------------------------------
extern "C" void kernel_launch(void* const* d_in, const int* in_sizes, int n_in,
                              void* d_out, int out_size, void* d_ws, size_t ws_size,
                              hipStream_t stream)
{
    const float* img_feat = (const float*)d_in[0];
    const float* option   = (const float*)d_in[1];
    const float* semantic = (const float*)d_in[2];
    const float* queue    = (const float*)d_in[3];
    const float* vae_w1 = (const float*)d_in[4];  const float* vae_b1 = (const float*)d_in[5];
    const float* vae_g1 = (const float*)d_in[6];  const float* vae_be1 = (const float*)d_in[7];
    const float* vae_w2 = (const float*)d_in[8];  const float* vae_b2 = (const float*)d_in[9];
    const float* opt_w1 = (const float*)d_in[10]; const float* opt_b1 = (const float*)d_in[11];
    const float* opt_g1 = (const float*)d_in[12]; const float* opt_be1 = (const float*)d_in[13];
    const float* opt_w2 = (const float*)d_in[14]; const float* opt_b2 = (const float*)d_in[15];
    const float* opt_g2 = (const float*)d_in[16]; const float* opt_be2 = (const float*)d_in[17];
    const float* sem_w1 = (const float*)d_in[18]; const float* sem_b1 = (const float*)d_in[19];
    const float* sem_g1 = (const float*)d_in[20]; const float* sem_be1 = (const float*)d_in[21];
    const float* sem_w2 = (const float*)d_in[22]; const float* sem_b2 = (const float*)d_in[23];
    const float* sem_g2 = (const float*)d_in[24]; const float* sem_be2 = (const float*)d_in[25];
    const float* fus_w  = (const float*)d_in[26]; const float* fus_b  = (const float*)d_in[27];
    const float* fus_g  = (const float*)d_in[28]; const float* fus_be = (const float*)d_in[29];
    const float* fus_sg = (const float*)d_in[30]; const float* fus_sb = (const float*)d_in[31];
    const float* cls_w1 = (const float*)d_in[32]; const float* cls_b1 = (const float*)d_in[33];
    const float* cls_g  = (const float*)d_in[34]; const float* cls_be = (const float*)d_in[35];
    const float* cls_w2 = (const float*)d_in[36]; const float* cls_b2 = (const float*)d_in[37];

    float* out = (float*)d_out;

    // ---- workspace carve-up (floats) ----
    float* ws = (float*)d_ws;
    size_t o = 0;
    auto alloc = [&](size_t n) { float* p = ws + o; o += n; return p; };
    float* tmp_pre = alloc((size_t)BATCH * DIM);   // pre-BN scratch
    float* hbuf    = alloc((size_t)BATCH * DIM);   // post-BN scratch
    float* img     = alloc((size_t)BATCH * DIM);
    float* optf    = alloc((size_t)BATCH * DIM);
    float* semf    = alloc((size_t)BATCH * DIM);
    float* fusedb  = alloc((size_t)BATCH * DIM);
    float* neigh   = alloc((size_t)BATCH * DIM);
    float* t192    = alloc((size_t)BATCH * 192);
    float* mean    = alloc(DIM);
    float* rstd    = alloc(DIM);
    float* fb      = alloc(4 * DIM);
    float* kn      = alloc(QN);
    float* fn      = alloc(BATCH);
    float* cand_v  = alloc((size_t)BATCH * 32 * KNN);
    int*   cand_i  = (int*)alloc((size_t)BATCH * 32 * KNN);

    auto gemm = [&](const float* A, const float* W, const float* bias, float* C,
                    int M, int N, int Kd, int accum) {
        gemm_wmma<<<dim3(N / 32, M / 16), 32, 0, stream>>>(A, W, bias, C, M, N, Kd, accum);
    };
    auto bnrelu = [&](const float* X, float* Y, const float* g, const float* be,
                      int M, int N) {
        bn_stats<<<N, 256, 0, stream>>>(X, M, N, mean, rstd);
        bn_apply_relu<<<(M * N + 255) / 256, 256, 0, stream>>>(X, Y, mean, rstd, g, be, M * N, N);
    };
    auto classifier = [&](const float* h, float* outp) {
        gemm(h, cls_w1, cls_b1, tmp_pre, BATCH, DIM, DIM, 0);
        bnrelu(tmp_pre, hbuf, cls_g, cls_be, BATCH, DIM);
        gemm_small<<<(BATCH * NCLS + 255) / 256, 256, 0, stream>>>(
            hbuf, cls_w2, cls_b2, outp, BATCH, NCLS, DIM);
    };

    // ---- img encoder: 2048 -> 512 (BN,ReLU) -> 512 ----
    gemm(img_feat, vae_w1, vae_b1, tmp_pre, BATCH, DIM, 2048, 0);
    bnrelu(tmp_pre, tmp_pre, vae_g1, vae_be1, BATCH, DIM);
    gemm(tmp_pre, vae_w2, vae_b2, img, BATCH, DIM, DIM, 0);

    // ---- option encoder: 768 -> 192 (BN,ReLU) -> 512 (BN,ReLU) ----
    gemm(option, opt_w1, opt_b1, t192, BATCH, 192, 768, 0);
    bnrelu(t192, t192, opt_g1, opt_be1, BATCH, 192);
    gemm(t192, opt_w2, opt_b2, tmp_pre, BATCH, DIM, 192, 0);
    bnrelu(tmp_pre, optf, opt_g2, opt_be2, BATCH, DIM);

    // ---- semantic encoder ----
    gemm(semantic, sem_w1, sem_b1, t192, BATCH, 192, 768, 0);
    bnrelu(t192, t192, sem_g1, sem_be1, BATCH, 192);
    gemm(t192, sem_w2, sem_b2, tmp_pre, BATCH, DIM, 192, 0);
    bnrelu(tmp_pre, semf, sem_g2, sem_be2, BATCH, DIM);

    // ---- fusion biases (fold indicator rows of fus_w into bias) ----
    fuse_bias<<<2, 256, 0, stream>>>(fus_w, fus_b, fb);

    // ---- fused = ReLU(BN(img@Wf0 + opt@Wf1 + sem@Wf2 + fb3)) ----
    gemm(img,  fus_w,                         fb + 3 * DIM, tmp_pre, BATCH, DIM, DIM, 0);
    gemm(optf, fus_w + (size_t)DIM * DIM,     nullptr,      tmp_pre, BATCH, DIM, DIM, 1);
    gemm(semf, fus_w + (size_t)2 * DIM * DIM, nullptr,      tmp_pre, BATCH, DIM, DIM, 1);
    bnrelu(tmp_pre, fusedb, fus_g, fus_be, BATCH, DIM);

    // ---- per-modality heads 0..2 ----
    const float* feats[3] = { img, optf, semf };
    for (int i = 0; i < 3; ++i) {
        gemm(feats[i], fus_w + (size_t)i * DIM * DIM, fb + i * DIM, tmp_pre, BATCH, DIM, DIM, 0);
        bnrelu(tmp_pre, hbuf, fus_sg + i * DIM, fus_sb + i * DIM, BATCH, DIM);
        classifier(hbuf, out + (size_t)i * BATCH * NCLS);
    }

    // ---- head 3: classifier(fused) ----
    classifier(fusedb, out + (size_t)3 * BATCH * NCLS);

    // ---- kNN retrieval head ----
    row_sqnorm<<<QN, 128, 0, stream>>>(queue, kn, DIM);
    row_sqnorm<<<BATCH, 128, 0, stream>>>(fusedb, fn, DIM);
    knn_partial<<<dim3(BATCH / 16, 32), 32, 0, stream>>>(queue, fusedb, kn, fn, cand_v, cand_i);
    knn_finalize<<<BATCH, 128, 0, stream>>>(cand_v, cand_i, queue, neigh);
    classifier(neigh, out + (size_t)4 * BATCH * NCLS);
}